// PredictionHead1D_82025285419603
// MI455X (gfx1250) — compile-verified
//
#include <hip/hip_runtime.h>
#include <math.h>

// gfx1250 / CDNA5: wave32, WMMA 16x16 shapes.
typedef __attribute__((ext_vector_type(2))) float        v2f;
typedef __attribute__((ext_vector_type(8))) float        v8f;
typedef __attribute__((ext_vector_type(4))) unsigned int v4u;
typedef __attribute__((ext_vector_type(8))) int          v8i;
typedef __attribute__((ext_vector_type(4))) int          v4i;

#define SEG_T 0.7f
#define GAU_T 0.7f
#define EPS_F 1e-7f

#define W80    80
#define NPIX   6400     // 80*80 pixels (= centers) per batch item
#define PTILES 400      // 16-pixel row tiles per batch item

// ---- Tensor Data Mover: 1D global->LDS DMA of `ndw` dwords.
// D# per CDNA5 ISA ch.8: group0 = {count/type/addr}, group1 = {dims/tile}.
// All fields are block-uniform so the descriptor stays in SGPRs.
// This toolchain exposes the 6-arg builtin:
//   (uint32x4 g0, int32x8 g1, int32x4 g2, int32x4 g3, int32x8 extra, i32 cpol)
__device__ static inline void tdm_load_1d(unsigned int lds_off,
                                          const void* gptr,
                                          unsigned int ndw) {
    const unsigned long long ga = (unsigned long long)(uintptr_t)gptr;
    v4u g0;
    g0[0] = 1u;                                         // count=1 (valid user D#)
    g0[1] = lds_off;                                    // lds_addr (bytes)
    g0[2] = (unsigned int)ga;                           // global_addr[31:0]
    g0[3] = (unsigned int)((ga >> 32) & 0x1FFFFFFu)     // global_addr[56:32]
          | (2u << 30);                                 // type=2 ("image")
    v8i g1;
    g1[0] = (int)(2u << 16);                            // data_size=2 (4 bytes)
    g1[1] = (int)((ndw & 0xFFFFu) << 16);               // tensor_dim0[15:0]
    g1[2] = (int)(((ndw >> 16) & 0xFFFFu) | (1u << 16));// tensor_dim0[31:16] | tensor_dim1=1
    g1[3] = (int)((ndw & 0xFFFFu) << 16);               // tile_dim0 = ndw
    g1[4] = 1;                                          // tile_dim1=1, tile_dim2=0
    g1[5] = (int)ndw;                                   // tensor_dim0_stride lo
    g1[6] = 0;                                          // stride0 hi | stride1 lo
    g1[7] = 0;                                          // stride1 hi
    const v4i z4 = {0, 0, 0, 0};                        // groups 2/3 unused (2D)
    const v8i z8 = {0, 0, 0, 0, 0, 0, 0, 0};
    __builtin_amdgcn_tensor_load_to_lds(g0, g1, z4, z4, z8, 0);
}

// One block = one (batch, center-row chunk) x 8 pixel tiles (8 waves).
// d2 tile (16 pixels x 16 centers) = ONE f32 WMMA with K=4:
//   A_row(p) = [pr, pc, pr^2+pc^2, 1], B_col(n) = [-2cr, -2cc, 1, cr^2+cc^2]
//   => (A.B)[p][n] = (pr-cr)^2 + (pc-cc)^2  (integer-exact in f32)
// exp() is monotonic => max-reduce s = -d2/denom[n] (masked), defer exp.
__global__ __launch_bounds__(256) void gauss_score_kernel(
    const float* __restrict__ var_map,
    const float* __restrict__ seg_map,
    float* __restrict__ dst,        // [S][B*NPIX] partials, or [B*NPIX] final
    int nBatch, int nSplit, int applyFinal)
{
    __shared__ float lds_v[NPIX];   // worst case (nSplit==1): 25.6 KB
    __shared__ float lds_s[NPIX];   // of the WGP's 320 KB LDS

    const int lane = threadIdx.x & 31;
    const int wid  = threadIdx.x >> 5;

    // blockIdx -> (chunk sc, batch b, pixel-tile group ptg)
    const int ptg = blockIdx.x % (PTILES / 8);
    const int tmp = blockIdx.x / (PTILES / 8);
    const int b   = tmp % nBatch;
    const int sc  = tmp / nBatch;
    const int pt  = ptg * 8 + wid;               // this wave's pixel tile

    const int rowsPerChunk = W80 / nSplit;
    const int chunkElems   = rowsPerChunk * W80; // dwords per plane chunk
    const int r0   = sc * rowsPerChunk;
    const int cOff = r0 * W80;

    const float* __restrict__ vb = var_map + b * NPIX;
    const float* __restrict__ sb = seg_map + b * NPIX;

    // ---- Stage the block-shared center chunk into LDS with the TDM.
    if (wid == 0) {
        tdm_load_1d((unsigned int)(uintptr_t)(void*)lds_v, vb + cOff,
                    (unsigned int)chunkElems);
        tdm_load_1d((unsigned int)(uintptr_t)(void*)lds_s, sb + cOff,
                    (unsigned int)chunkElems);
        __builtin_amdgcn_s_wait_tensorcnt(0);    // DMA done before barrier
    }
    __syncthreads();

    const int  nl = lane & 15;
    const bool hi = lane >= 16;

    // ---- Loop-invariant A (16x4 f32); ISA 32-bit A striping:
    // lanes 0-15: VGPR0=K0, VGPR1=K1 ; lanes 16-31: VGPR0=K2, VGPR1=K3.
    const int   p  = pt * 16 + nl;
    const float pr = (float)(p / W80);
    const float pc = (float)(p % W80);
    v2f A;
    A.x = hi ? (pr * pr + pc * pc) : pr;    // K0=pr | K2=pr^2+pc^2
    A.y = hi ? 1.0f                : pc;    // K1=pc | K3=1

    // Per-lane center-column invariants (80 cols = 5 tiles of 16).
    float m2cc[5], cc2[5];
#pragma unroll
    for (int s = 0; s < 5; ++s) {
        const float cc = (float)(nl + 16 * s);
        m2cc[s] = -2.0f * cc;
        cc2[s]  = cc * cc;
    }

    float maxacc[8];
#pragma unroll
    for (int i = 0; i < 8; ++i) maxacc[i] = -__builtin_inff();

    float fcr = (float)r0;
    int   rowBase = nl;                      // LDS index of this lane's column

    for (int r = 0; r < rowsPerChunk; ++r) {
        const float m2cr = -2.0f * fcr;
        const float cr2  = fcr * fcr;
        const float Bx   = hi ? 1.0f : m2cr; // shared by all 5 slots

#pragma unroll
        for (int s = 0; s < 5; ++s) {
            const float v = lds_v[rowBase + 16 * s];   // ds_load_b32 (broadcast
            const float g = lds_s[rowBase + 16 * s];   //  across half-waves)
            const float t = v + EPS_F;
            // -1/(2 t^2): single v_rcp_f32 (TRANS, co-executes with WMMA).
            const float ninvd = -0.5f * __builtin_amdgcn_rcpf(t * t);
            // Fold selection into the scale: unselected => d2*(-inf) = -inf
            // (d2>0) or NaN (the single d2==0 pair); v_max_num drops NaN.
            const float scale = (g > SEG_T) ? ninvd : -__builtin_inff();

            v2f B;                               // B (4x16), same K striping
            B.x = Bx;                            // K0=-2cr | K2=1
            B.y = hi ? (cr2 + cc2[s]) : m2cc[s]; // K1=-2cc | K3=cr^2+cc^2

            v8f C = {};
            // 8 args: (neg_a, A, neg_b, B, c_mod, C, reuse_a, reuse_b)
            C = __builtin_amdgcn_wmma_f32_16x16x4_f32(
                    false, A, false, B, (short)0, C, false, false);

#pragma unroll
            for (int i = 0; i < 8; ++i)          // element i: (M=i+8*hi, N=nl)
                maxacc[i] = fmaxf(maxacc[i], C[i] * scale);
        }
        fcr += 1.0f;
        rowBase += W80;
    }

    // Per-row max across the 16 columns of each half-wave (xor 1,2,4,8 keeps
    // bit4, so rows 0-7 / 8-15 reduce independently, matching the C layout).
#pragma unroll
    for (int i = 0; i < 8; ++i) {
        float m = maxacc[i];
        m = fmaxf(m, __shfl_xor(m, 1));
        m = fmaxf(m, __shfl_xor(m, 2));
        m = fmaxf(m, __shfl_xor(m, 4));
        m = fmaxf(m, __shfl_xor(m, 8));
        maxacc[i] = m;
    }

    float res[8];
    if (applyFinal) {   // direct path (nSplit==1): one exp per pixel + threshold
#pragma unroll
        for (int i = 0; i < 8; ++i) {
            const float gg = __expf(maxacc[i] + EPS_F);   // exp(-inf)=0
            res[i] = (gg >= GAU_T) ? gg : 0.0f;
        }
    } else {            // partial path: raw scores, finalize kernel combines
#pragma unroll
        for (int i = 0; i < 8; ++i) res[i] = maxacc[i];
    }

    float* __restrict__ dp = dst + (size_t)(sc * nBatch + b) * NPIX + pt * 16;
    if (lane == 0) {
#pragma unroll
        for (int i = 0; i < 8; ++i) dp[i] = res[i];       // rows 0-7
    } else if (lane == 16) {
#pragma unroll
        for (int i = 0; i < 8; ++i) dp[8 + i] = res[i];   // rows 8-15
    }
}

// Combine the S partial score planes: max, one exp, threshold.
__global__ __launch_bounds__(256) void gauss_finalize_kernel(
    const float* __restrict__ part, float* __restrict__ out,
    int nTotal, int nSplit)
{
    const int i = blockIdx.x * blockDim.x + threadIdx.x;
    if (i >= nTotal) return;
    float m = -__builtin_inff();
    for (int s = 0; s < nSplit; ++s)
        m = fmaxf(m, part[(size_t)s * nTotal + i]);
    const float g = __expf(m + EPS_F);
    out[i] = (g >= GAU_T) ? g : 0.0f;
}

extern "C" void kernel_launch(void* const* d_in, const int* in_sizes, int n_in,
                              void* d_out, int out_size, void* d_ws, size_t ws_size,
                              hipStream_t stream) {
    const float* var_map = (const float*)d_in[0];   // (B,1,80,80) f32
    const float* seg_map = (const float*)d_in[1];   // (B,1,80,80) f32
    float* out = (float*)d_out;                     // (B,1,80,80) f32

    const int B = in_sizes[0] / NPIX;               // 2 in the reference
    const size_t plane = (size_t)B * NPIX * sizeof(float);

    // Largest center-row split whose partials fit in d_ws.
    int S = 1;
    if      (ws_size >= 8 * plane) S = 8;
    else if (ws_size >= 4 * plane) S = 4;
    else if (ws_size >= 2 * plane) S = 2;

    const int blocks = (PTILES / 8) * B * S;        // 8 pixel tiles per block

    if (S == 1) {
        // Direct single-pass path (100 blocks, 800 waves).
        gauss_score_kernel<<<blocks, 256, 0, stream>>>(
            var_map, seg_map, out, B, 1, 1);
    } else {
        // Sharded path: 6400 waves at S=8 to fill the MI455X's SIMDs,
        // then a tiny deterministic combine.
        gauss_score_kernel<<<blocks, 256, 0, stream>>>(
            var_map, seg_map, (float*)d_ws, B, S, 0);
        const int nTotal = B * NPIX;
        gauss_finalize_kernel<<<(nTotal + 255) / 256, 256, 0, stream>>>(
            (const float*)d_ws, out, nTotal, S);
    }
}